// SpatialBilinearUpsample_35665408426065
// MI455X (gfx1250) — compile-verified
//
#include <hip/hip_runtime.h>

typedef __attribute__((ext_vector_type(2))) float v2f;
typedef __attribute__((ext_vector_type(8))) float v8f;

#define IN_H 128
#define IN_W 128
#define OUT_H 256
#define OUT_W 256
#define PLANES 1024          // 2*64*8
#define TILES_PER_PLANE 128  // (256/32) row-blocks * (256/16) col-blocks
#define WAVES_PER_BLOCK 4

// Weight of (local) input sample r contributing to (local) output sample o,
// for 2x bilinear with taps {0.25, 0.75}. Valid when the tile bases are even
// (out-row base is a multiple of 32, out-col base a multiple of 16).
//   o even: 0.25 @ r=o/2,      0.75 @ r=o/2+1
//   o odd : 0.75 @ r=(o+1)/2,  0.25 @ r=(o+1)/2+1
__device__ __forceinline__ float wgt(int o, int r) {
  int lo = (o + 1) >> 1;
  float wlo = (o & 1) ? 0.75f : 0.25f;
  float whi = 1.0f - wlo;
  return (r == lo) ? wlo : ((r == lo + 1) ? whi : 0.0f);
}

__global__ __launch_bounds__(128) void SpatialBilinearUpsample_kernel(
    const float* __restrict__ x, float* __restrict__ out) {
  // Per-wave LDS slices.
  // s_x: input halo region, 18 rows x 10 cols, padded to 32x12 and zeroed
  //      (rows 18..31 / cols 10..11 are the M/K padding for WMMA).
  // s_h: horizontally-upsampled rows, 18 rows x 16 out-cols, padded to 32x16.
  __shared__ float s_x[WAVES_PER_BLOCK][32 * 12];
  __shared__ float s_h[WAVES_PER_BLOCK][32 * 16];

  const int lane = threadIdx.x & 31;
  const int wv   = threadIdx.x >> 5;
  const int tid  = blockIdx.x * WAVES_PER_BLOCK + wv;   // tile id, grid sized exactly

  const int p      = tid >> 7;        // plane 0..1023
  const int rem    = tid & 127;
  const int rowblk = rem >> 4;        // 0..7   -> 32 output rows each
  const int colblk = rem & 15;        // 0..15  -> 16 output cols each
  const int ibase  = rowblk * 16;     // input row base of tile
  const int obase  = rowblk * 32;     // output row base
  const int jbase  = colblk * 8;      // input col base
  const int ocbase = colblk * 16;     // output col base

  float* Xl = s_x[wv];
  float* Hl = s_h[wv];

  // Zero staging buffer, then load the clamped 18x10 input halo region.
  for (int i = lane; i < 32 * 12; i += 32) Xl[i] = 0.0f;
  const float* xp = x + (size_t)p * (IN_H * IN_W);
  for (int i = lane; i < 18 * 10; i += 32) {
    int r = i / 10, c = i % 10;
    int gr = min(max(ibase - 1 + r, 0), IN_H - 1);   // replicate pad == clamp
    int gc = min(max(jbase - 1 + c, 0), IN_W - 1);
    Xl[r * 12 + c] = xp[gr * IN_W + gc];
  }
  __syncthreads();

  const int n    = lane & 15;   // N index (and M index for A-operand reads)
  const int half = lane >> 4;   // 0: K{0,1} / rows 0-7 of D; 1: K{2,3} / rows 8-15
  const int koff = half * 2;

  // ---- Horizontal pass: H[r][n] = sum_c X[r][c] * wh(n, c) ----
  // A = X (M=region rows, K=region cols), B = weights.
  for (int mb = 0; mb < 2; ++mb) {
    v8f d = {};
    for (int kc = 0; kc < 3; ++kc) {
      int k0 = kc * 4 + koff;
      int m  = mb * 16 + n;
      v2f a, b;
      a.x = Xl[m * 12 + k0];
      a.y = Xl[m * 12 + k0 + 1];
      b.x = wgt(n, k0);
      b.y = wgt(n, k0 + 1);
      d = __builtin_amdgcn_wmma_f32_16x16x4_f32(false, a, false, b,
                                                (short)0, d, false, false);
    }
    for (int i = 0; i < 8; ++i) {
      int row = mb * 16 + half * 8 + i;
      Hl[row * 16 + n] = d[i];
    }
  }
  __syncthreads();

  // ---- Vertical pass: Out[m][n] = sum_r wv(m, r) * H[r][n] ----
  // A = weights (M=out rows, K=H rows), B = H data.
  float* op = out + (size_t)p * (OUT_H * OUT_W);
  for (int mb = 0; mb < 2; ++mb) {
    v8f d = {};
    for (int kc = 0; kc < 5; ++kc) {
      int k0 = kc * 4 + koff;
      int m  = mb * 16 + n;
      v2f a, b;
      a.x = wgt(m, k0);
      a.y = wgt(m, k0 + 1);
      b.x = Hl[k0 * 16 + n];
      b.y = Hl[(k0 + 1) * 16 + n];
      d = __builtin_amdgcn_wmma_f32_16x16x4_f32(false, a, false, b,
                                                (short)0, d, false, false);
    }
    for (int i = 0; i < 8; ++i) {
      int row = obase + mb * 16 + half * 8 + i;
      op[(size_t)row * OUT_W + ocbase + n] = d[i];
    }
  }
}

extern "C" void kernel_launch(void* const* d_in, const int* in_sizes, int n_in,
                              void* d_out, int out_size, void* d_ws, size_t ws_size,
                              hipStream_t stream) {
  (void)in_sizes; (void)n_in; (void)out_size; (void)d_ws; (void)ws_size;
  const float* x = (const float*)d_in[0];   // filt (d_in[1]) is baked into the kernel
  float* out = (float*)d_out;
  const int total_tiles = PLANES * TILES_PER_PLANE;               // 131072
  const int blocks = total_tiles / WAVES_PER_BLOCK;               // 32768
  SpatialBilinearUpsample_kernel<<<blocks, 32 * WAVES_PER_BLOCK, 0, stream>>>(x, out);
}